// MaxUnpool2D_59047210385945
// MI455X (gfx1250) — compile-verified
//
#include <hip/hip_runtime.h>

// MaxUnpool2D, B=16 H=64 W=64 C=128, 2x2 windows -> out [16,128,128,128].
//
// Scatter inverted into window-owned writes: each input element (b,h,w,c)
// owns the 4 output cells of its 2x2 window at channel c. Write the matching
// cell with the value, the other 3 with zero. Every output element is written
// exactly once -> no zero-init pass, no atomics, all accesses 128-bit and
// contiguous per wave32. Traffic = 64 MB read + 128 MB write = HBM roofline
// (~192 MB / 23.3 TB/s ~= 8.3 us).

typedef float vf4 __attribute__((ext_vector_type(4)));
typedef int   vi4 __attribute__((ext_vector_type(4)));

static __device__ __forceinline__ float sel(int m, int e, float v) {
    return (m == e) ? v : 0.0f;
}

__global__ __launch_bounds__(256) void maxunpool2d_kernel(
    const float* __restrict__ in,    // [16,64,64,128]
    const int*   __restrict__ mask,  // [16,64,64,128], flat (ho*128+wo)*128+c
    float*       __restrict__ out)   // [16,128,128,128]
{
    const int gtid = blockIdx.x * 256 + threadIdx.x;
    const int site = gtid >> 5;          // (b,h,w) site; one wave32 per site
    const int lane = gtid & 31;
    const int c    = lane << 2;          // 4 channels per lane (128-bit ops)

    const int w = site & 63;
    const int h = (site >> 6) & 63;
    const int b = site >> 12;

    // Streaming 128-bit loads with non-temporal hint (read-once data).
    const long in_base = (long)site * 128 + c;
    const vf4 v = __builtin_nontemporal_load((const vf4*)(in + in_base));
    const vi4 m = __builtin_nontemporal_load((const vi4*)(mask + in_base));

    // Within-batch flat output index of window cell (dh=0,dw=0), channel c.
    // Ho=Wo=128, C=128: idx = ho*16384 + wo*128 + c. Mask values use the
    // same encoding (no batch), so matching is a single integer compare.
    const int  idx00 = (h << 15) + (w << 8) + c;   // (2h)*16384 + (2w)*128 + c
    const long obase = (long)b * (128 * 128 * 128);

#pragma unroll
    for (int dh = 0; dh < 2; ++dh) {
#pragma unroll
        for (int dw = 0; dw < 2; ++dw) {
            const int e = idx00 + dh * 16384 + dw * 128;
            vf4 o;
            o.x = sel(m.x, e + 0, v.x);
            o.y = sel(m.y, e + 1, v.y);
            o.z = sel(m.z, e + 2, v.z);
            o.w = sel(m.w, e + 3, v.w);
            // Non-temporal 128-bit store: gfx1250 TH=NT cache-control path,
            // streams 128 MB of output without evicting the L2 working set.
            __builtin_nontemporal_store(o, (vf4*)(out + obase + e));
        }
    }
}

extern "C" void kernel_launch(void* const* d_in, const int* in_sizes, int n_in,
                              void* d_out, int out_size, void* d_ws, size_t ws_size,
                              hipStream_t stream) {
    (void)in_sizes; (void)n_in; (void)d_ws; (void)ws_size; (void)out_size;
    const float* in   = (const float*)d_in[0];   // input_pool, float32
    const int*   mask = (const int*)d_in[1];     // pool_mask, int32
    float*       out  = (float*)d_out;           // float32 [16,128,128,128]

    // B*H*W*C / 4 channels-per-thread = 2,097,152 threads; 256/block -> 8192 blocks.
    const int threads = 256;
    const int blocks  = (16 * 64 * 64 * 128 / 4) / threads;
    maxunpool2d_kernel<<<blocks, threads, 0, stream>>>(in, mask, out);
}